// VectorQuantizer_45440753992252
// MI455X (gfx1250) — compile-verified
//
#include <hip/hip_runtime.h>
#include <hip/hip_bf16.h>

typedef __attribute__((ext_vector_type(2))) float v2f;
typedef __attribute__((ext_vector_type(8))) float v8f;

// Problem constants (from reference setup_inputs)
#define BATCH   16
#define TSEQ    1024
#define NROWS   (BATCH * TSEQ)   // 16384
#define DDIM    512
#define KCODES  2048

// GEMM-argmin tiling
#define STRIPS     2             // 16-row strips per wave (B-frag reuse)
#define KTILES     2             // concurrent 16-col k-tiles (A-frag reuse)
#define WAVES      2             // waves per workgroup
#define LDS_STRIDE 516           // padded row stride (floats): 516 % 64 == 4 -> conflict-free frag reads
#define ROWS_PER_WAVE  (STRIPS * 16)
#define ROWS_PER_BLOCK (WAVES * ROWS_PER_WAVE)

// ---------------------------------------------------------------------------
// Kernel 1: c2[k] = ||codebook[k]||^2   (one wave per codebook row)
// ---------------------------------------------------------------------------
__global__ void vq_rownorm_kernel(const float* __restrict__ cb, float* __restrict__ c2) {
    const int lane = threadIdx.x & 31;
    const int row  = blockIdx.x * (blockDim.x >> 5) + (threadIdx.x >> 5);
    const float4* p = (const float4*)(cb + (size_t)row * DDIM);
    float s = 0.0f;
    for (int i = lane; i < DDIM / 4; i += 32) {
        float4 v = p[i];
        s = fmaf(v.x, v.x, s);
        s = fmaf(v.y, v.y, s);
        s = fmaf(v.z, v.z, s);
        s = fmaf(v.w, v.w, s);
    }
#pragma unroll
    for (int off = 16; off > 0; off >>= 1) s += __shfl_xor(s, off, 32);
    if (lane == 0) c2[row] = s;
}

// ---------------------------------------------------------------------------
// Kernel 2: fused fp32-WMMA GEMM + argmin.
//   For each row i: nn[i] = argmin_k ( c2[k] - 2 * <x_i, c_k> )
// Each wave: 2 strips of 16 rows (A in LDS, loaded once) x 2 concurrent
// k-tiles (codebook streamed from L2). Each A frag feeds 2 wmmas, each B frag
// feeds 2 wmmas; 4 independent accumulator chains hide wmma latency.
// ---------------------------------------------------------------------------
__global__ void vq_gemm_argmin_kernel(const float* __restrict__ x,
                                      const float* __restrict__ cb,
                                      const float* __restrict__ c2,
                                      int* __restrict__ nn) {
    extern __shared__ float smem[];
    const int lane = threadIdx.x & 31;
    const int wave = threadIdx.x >> 5;
    const int half = lane >> 4;      // 0: lanes 0-15, 1: lanes 16-31
    const int l16  = lane & 15;

    const int waveRow0 = blockIdx.x * ROWS_PER_BLOCK + wave * ROWS_PER_WAVE;
    float* As = smem + wave * (ROWS_PER_WAVE * LDS_STRIDE);

    // ---- Stage A strips into LDS (coalesced float4 global loads) ----
    for (int j = lane; j < ROWS_PER_WAVE * (DDIM / 4); j += 32) {
        const int row = j >> 7;          // DDIM/4 = 128 float4 per row
        const int c4  = j & 127;
        const float4 v = ((const float4*)(x + (size_t)(waveRow0 + row) * DDIM))[c4];
        *(float4*)(As + row * LDS_STRIDE + c4 * 4) = v;
    }
    // Same-wave DS store->load is kept in order by DScnt; no cross-wave sharing.

    // ---- Running per-row argmin state (C/D layout: lane=col N, vgpr=row M) ----
    float minv[STRIPS][8];
    int   mini[STRIPS][8];
#pragma unroll
    for (int s = 0; s < STRIPS; ++s)
#pragma unroll
        for (int v = 0; v < 8; ++v) { minv[s][v] = 3.4e38f; mini[s][v] = 0; }

    // ---- k loop: KTILES*16 codebook entries per outer iteration ----
    for (int k0 = 0; k0 < KCODES; k0 += 16 * KTILES) {
        v8f acc[STRIPS][KTILES];
#pragma unroll
        for (int s = 0; s < STRIPS; ++s)
#pragma unroll
            for (int t = 0; t < KTILES; ++t)
                acc[s][t] = (v8f){0.f, 0.f, 0.f, 0.f, 0.f, 0.f, 0.f, 0.f};

        // B fragment bases: lane's column = codebook row (k0 + 16t + l16)
        const float* bbase = cb + (size_t)(k0 + l16) * DDIM + 2 * half;
#pragma unroll 4
        for (int d = 0; d < DDIM; d += 4) {
            v2f bfrag[KTILES];
#pragma unroll
            for (int t = 0; t < KTILES; ++t)
                bfrag[t] = *(const v2f*)(bbase + (size_t)(16 * t) * DDIM + d);
#pragma unroll
            for (int s = 0; s < STRIPS; ++s) {
                const v2f afrag = *(const v2f*)(As + (s * 16 + l16) * LDS_STRIDE + d + 2 * half);
#pragma unroll
                for (int t = 0; t < KTILES; ++t)
                    acc[s][t] = __builtin_amdgcn_wmma_f32_16x16x4_f32(
                        false, afrag, false, bfrag[t], (short)0, acc[s][t], false, false);
            }
        }

        // delta column for this lane: k = k0 + 16t + l16 ; delta = c2[k] - 2*xc
#pragma unroll
        for (int t = 0; t < KTILES; ++t) {
            const int   kidx = k0 + 16 * t + l16;
            const float c2v  = c2[kidx];
#pragma unroll
            for (int s = 0; s < STRIPS; ++s)
#pragma unroll
                for (int v = 0; v < 8; ++v) {
                    const float dv = fmaf(-2.0f, acc[s][t][v], c2v);
                    if (dv < minv[s][v]) { minv[s][v] = dv; mini[s][v] = kidx; }
                }
        }
    }

    // ---- Reduce across the 16 lanes of each half (rows 0-7 in lanes 0-15,
    //      rows 8-15 in lanes 16-31); xor-shuffle stays within each half ----
#pragma unroll
    for (int s = 0; s < STRIPS; ++s)
#pragma unroll
        for (int v = 0; v < 8; ++v) {
            float val = minv[s][v];
            int   idx = mini[s][v];
#pragma unroll
            for (int off = 8; off > 0; off >>= 1) {
                const float oval = __shfl_xor(val, off, 32);
                const int   oidx = __shfl_xor(idx, off, 32);
                if (oval < val || (oval == val && oidx < idx)) { val = oval; idx = oidx; }
            }
            if (l16 == 0)
                nn[waveRow0 + s * 16 + half * 8 + v] = idx;
        }
}

// ---------------------------------------------------------------------------
// Kernel 3: gather + STE + loss.
//   quantized_ste = codebook[nn[row]]   (numerically equals inputs + sg(q-x))
//   loss          = (q - x)^2
// ---------------------------------------------------------------------------
__global__ void vq_finalize_kernel(const float* __restrict__ x,
                                   const float* __restrict__ cb,
                                   const int* __restrict__ nn,
                                   float* __restrict__ quant,
                                   float* __restrict__ loss) {
    const size_t gid = (size_t)blockIdx.x * blockDim.x + threadIdx.x;
    if (gid >= (size_t)NROWS * (DDIM / 4)) return;
    const int row = (int)(gid >> 7);     // DDIM/4 = 128
    const int c4  = (int)(gid & 127);
    const int k   = nn[row];
    const float4 q  = ((const float4*)(cb + (size_t)k * DDIM))[c4];
    const float4 xv = ((const float4*)(x + (size_t)row * DDIM))[c4];
    ((float4*)(quant + (size_t)row * DDIM))[c4] = q;
    float4 l;
    l.x = (q.x - xv.x) * (q.x - xv.x);
    l.y = (q.y - xv.y) * (q.y - xv.y);
    l.z = (q.z - xv.z) * (q.z - xv.z);
    l.w = (q.w - xv.w) * (q.w - xv.w);
    ((float4*)(loss + (size_t)row * DDIM))[c4] = l;
}

// ---------------------------------------------------------------------------
extern "C" void kernel_launch(void* const* d_in, const int* in_sizes, int n_in,
                              void* d_out, int out_size, void* d_ws, size_t ws_size,
                              hipStream_t stream) {
    const float* x  = (const float*)d_in[0];   // [16,1024,512] f32
    const float* cb = (const float*)d_in[1];   // [2048,512]    f32

    float* quant = (float*)d_out;                                // [16,1024,512]
    float* loss  = quant + (size_t)NROWS * DDIM;                 // [16,1024,512]
    int*   nn    = (int*)(loss + (size_t)NROWS * DDIM);          // [1,16,1024]
    float* cbout = (float*)(nn + NROWS);                         // [1,2048,512]

    float* c2 = (float*)d_ws;                                    // [2048]

    // 1) codebook row norms: 8 waves/block, one row per wave
    vq_rownorm_kernel<<<KCODES / 8, 256, 0, stream>>>(cb, c2);

    // 2) fused fp32-WMMA GEMM + argmin
    const size_t ldsBytes = (size_t)WAVES * ROWS_PER_WAVE * LDS_STRIDE * sizeof(float);
    vq_gemm_argmin_kernel<<<NROWS / ROWS_PER_BLOCK, WAVES * 32, ldsBytes, stream>>>(
        x, cb, c2, nn);

    // 3) gather + loss
    const size_t nthreads = (size_t)NROWS * (DDIM / 4);
    vq_finalize_kernel<<<(int)((nthreads + 255) / 256), 256, 0, stream>>>(
        x, cb, nn, quant, loss);

    // 4) codebook passthrough output
    hipMemcpyAsync(cbout, cb, (size_t)KCODES * DDIM * sizeof(float),
                   hipMemcpyDeviceToDevice, stream);
}